// EquivariantModel_1563368095921
// MI455X (gfx1250) — compile-verified
//
#include <hip/hip_runtime.h>
#include <math.h>

typedef __attribute__((ext_vector_type(2))) float v2f;
typedef __attribute__((ext_vector_type(8))) float v8f;

#if defined(__has_builtin)
#if __has_builtin(__builtin_amdgcn_wmma_f32_16x16x4_f32)
#define HAVE_WMMA_F32X4 1
#endif
#endif

// ---------------------------------------------------------------------------
// Prep kernel (1 block, 32 threads): fold all weight-only algebra into
// alpha[32], beta[32] so that  hid[b,o] = relu(alpha[o] + T[b]*beta[o]).
//   h0[c]  = (tp_w0 . lin_w0[:,c]) / sqrt(8)        -> inv0[c] = 36*h0[c]^2
//   w1v[v] = (tp_w1 . lin_w1[:,v]) / 2              -> inv1[v] = 3*w1v[v]^2 * T[b]
//   alpha[o] = clf_b1[o] + sum_c inv0[c]*clf_w1[c,o]
//   beta[o]  = sum_v 3*w1v[v]^2 * clf_w1[8+v,o]
// ---------------------------------------------------------------------------
__global__ void eq_prep_kernel(const float* __restrict__ tp_w0,
                               const float* __restrict__ tp_w1,
                               const float* __restrict__ lin_w0,
                               const float* __restrict__ lin_w1,
                               const float* __restrict__ clf_w1,
                               const float* __restrict__ clf_b1,
                               float* __restrict__ ws) {
  int o = threadIdx.x;  // 0..31
  const float is8 = 0.35355339059327373f;  // 1/sqrt(8)
  float alpha = clf_b1[o];
#pragma unroll
  for (int c = 0; c < 8; ++c) {
    float h = 0.f;
#pragma unroll
    for (int k = 0; k < 8; ++k) h += tp_w0[k] * lin_w0[k * 8 + c];
    h *= is8;
    alpha += 36.0f * h * h * clf_w1[c * 32 + o];
  }
  float beta = 0.f;
#pragma unroll
  for (int v = 0; v < 4; ++v) {
    float w = 0.f;
#pragma unroll
    for (int u = 0; u < 4; ++u) w += tp_w1[u] * lin_w1[u * 4 + v];
    w *= 0.5f;
    beta += 3.0f * w * w * clf_w1[(8 + v) * 32 + o];
  }
  ws[o] = alpha;
  ws[32 + o] = beta;
}

__device__ __forceinline__ void unitv(float ax, float ay, float az,
                                      float bx, float by, float bz,
                                      float& ux, float& uy, float& uz) {
  float dx = bx - ax, dy = by - ay, dz = bz - az;
  float inv = 1.0f / sqrtf(dx * dx + dy * dy + dz * dz);
  ux = dx * inv; uy = dy * inv; uz = dz * inv;
}

// ---------------------------------------------------------------------------
// Main fused kernel: one lane per batch row computes T[b]; each wave then
// forms Hid(16x32) tiles and runs the 32->2 classifier head as 8x
// V_WMMA_F32_16X16X4_F32 against zero-padded clf_w2 (32x16). Results are
// staged through LDS so every lane emits one coalesced float2 store.
// ---------------------------------------------------------------------------
__global__ __launch_bounds__(256) void eq_main_kernel(
    const float* __restrict__ pos, const float* __restrict__ wsAB,
    const float* __restrict__ clf_w2, const float* __restrict__ clf_b2,
    float* __restrict__ out, int Btot) {
  const int lane = threadIdx.x & 31;
  const int wave = threadIdx.x >> 5;
  int r = blockIdx.x * 256 + threadIdx.x;
  int rc = r < Btot ? r : Btot - 1;  // clamp (keeps EXEC full for WMMA)

  // coalesced 48B/lane load of the 4 node positions
  const float4* p4 = (const float4*)(pos + (size_t)rc * 12);
  float4 q0 = p4[0], q1 = p4[1], q2 = p4[2];
  float p0x = q0.x, p0y = q0.y, p0z = q0.z;
  float p1x = q0.w, p1y = q1.x, p1z = q1.y;
  float p2x = q1.z, p2y = q1.w, p2z = q2.x;
  float p3x = q2.y, p3y = q2.z, p3z = q2.w;

  // 6 unique unit direction vectors
  float u01x,u01y,u01z, u02x,u02y,u02z, u03x,u03y,u03z;
  float u12x,u12y,u12z, u13x,u13y,u13z, u23x,u23y,u23z;
  unitv(p0x,p0y,p0z, p1x,p1y,p1z, u01x,u01y,u01z);
  unitv(p0x,p0y,p0z, p2x,p2y,p2z, u02x,u02y,u02z);
  unitv(p0x,p0y,p0z, p3x,p3y,p3z, u03x,u03y,u03z);
  unitv(p1x,p1y,p1z, p2x,p2y,p2z, u12x,u12y,u12z);
  unitv(p1x,p1y,p1z, p3x,p3y,p3z, u13x,u13y,u13z);
  unitv(p2x,p2y,p2z, p3x,p3y,p3z, u23x,u23y,u23z);

  float s0x =  u01x + u02x + u03x, s0y =  u01y + u02y + u03y, s0z =  u01z + u02z + u03z;
  float s1x = -u01x + u12x + u13x, s1y = -u01y + u12y + u13y, s1z = -u01z + u12z + u13z;
  float s2x = -u02x - u12x + u23x, s2y = -u02y - u12y + u23y, s2z = -u02z - u12z + u23z;
  float s3x = -u03x - u13x - u23x, s3y = -u03y - u13y - u23y, s3z = -u03z - u13z - u23z;
  float T = s0x*s0x + s0y*s0y + s0z*s0z + s1x*s1x + s1y*s1y + s1z*s1z +
            s2x*s2x + s2y*s2y + s2z*s2z + s3x*s3x + s3y*s3y + s3z*s3z;

#if HAVE_WMMA_F32X4
  __shared__ float ldsOut[512];  // 8 waves * 32 rows * 2 cols

  // Per-lane slice of alpha/beta matching the 16x4 f32 A layout:
  // lanes 0-15 supply (K=4k, 4k+1), lanes 16-31 supply (K=4k+2, 4k+3).
  const int off = (lane < 16) ? 0 : 2;
  const int n = lane & 15;
  float aC[16], bC[16];
#pragma unroll
  for (int j = 0; j < 16; ++j) {
    int col = 4 * (j >> 1) + off + (j & 1);
    aC[j] = wsAB[col];
    bC[j] = wsAB[32 + col];
  }
  // B fragments: zero-padded clf_w2 (32x16), rows K striped like A.
  v2f bfr[8];
#pragma unroll
  for (int k = 0; k < 8; ++k) {
    int k0 = 4 * k + off;
    float b0 = clf_w2[k0 * 2 + (n & 1)];
    float b1 = clf_w2[(k0 + 1) * 2 + (n & 1)];
    bfr[k].x = (n < 2) ? b0 : 0.0f;
    bfr[k].y = (n < 2) ? b1 : 0.0f;
  }

#pragma unroll
  for (int t = 0; t < 2; ++t) {
    float Trow = __shfl(T, t * 16 + (lane & 15), 32);
    v8f c = {};
#pragma unroll
    for (int k = 0; k < 8; ++k) {
      float h0v = aC[2 * k]     + Trow * bC[2 * k];
      float h1v = aC[2 * k + 1] + Trow * bC[2 * k + 1];
      v2f a;
      a.x = h0v > 0.f ? h0v : 0.f;
      a.y = h1v > 0.f ? h1v : 0.f;
      c = __builtin_amdgcn_wmma_f32_16x16x4_f32(false, a, false, bfr[k],
                                                (short)0, c, false, false);
    }
    // D layout: lane L -> N = L&15, rows M = (L<16?0:8)+v. Only N in {0,1}.
    // Stage the two valid columns into LDS for a coalesced final store.
    if (n < 2) {
      int mBase = (lane < 16) ? 0 : 8;
      int base = wave * 64 + (t * 16 + mBase) * 2 + n;
#pragma unroll
      for (int v = 0; v < 8; ++v) ldsOut[base + v * 2] = c[v];
    }
  }
  __syncthreads();
  if (r < Btot) {
    float2 o2;
    o2.x = ldsOut[(wave * 32 + lane) * 2 + 0] + clf_b2[0];
    o2.y = ldsOut[(wave * 32 + lane) * 2 + 1] + clf_b2[1];
    ((float2*)out)[r] = o2;  // 256B contiguous per wave
  }
#else
  // VALU fallback: per-lane classifier head.
  float o0 = clf_b2[0], o1 = clf_b2[1];
#pragma unroll
  for (int k = 0; k < 32; ++k) {
    float h = wsAB[k] + T * wsAB[32 + k];
    h = h > 0.f ? h : 0.f;
    o0 += h * clf_w2[k * 2 + 0];
    o1 += h * clf_w2[k * 2 + 1];
  }
  if (r < Btot) {
    out[r * 2 + 0] = o0;
    out[r * 2 + 1] = o1;
  }
#endif
}

extern "C" void kernel_launch(void* const* d_in, const int* in_sizes, int n_in,
                              void* d_out, int out_size, void* d_ws, size_t ws_size,
                              hipStream_t stream) {
  const float* pos    = (const float*)d_in[0];
  const float* tp_w0  = (const float*)d_in[1];
  const float* tp_w1  = (const float*)d_in[2];
  const float* lin_w0 = (const float*)d_in[3];
  const float* lin_w1 = (const float*)d_in[4];
  const float* clf_w1 = (const float*)d_in[5];
  const float* clf_b1 = (const float*)d_in[6];
  const float* clf_w2 = (const float*)d_in[7];
  const float* clf_b2 = (const float*)d_in[8];
  float* out = (float*)d_out;
  float* ws  = (float*)d_ws;

  int B = in_sizes[0] / 12;  // pos is [B,4,3]

  hipLaunchKernelGGL(eq_prep_kernel, dim3(1), dim3(32), 0, stream,
                     tp_w0, tp_w1, lin_w0, lin_w1, clf_w1, clf_b1, ws);

  int grid = (B + 255) / 256;
  hipLaunchKernelGGL(eq_main_kernel, dim3(grid), dim3(256), 0, stream,
                     pos, ws, clf_w2, clf_b2, out, B);
}